// ChunkedRichPoolMLP_83880711291219
// MI455X (gfx1250) — compile-verified
//
#include <hip/hip_runtime.h>
#include <math.h>

// Problem constants (match reference)
#define BATCH   64
#define TLEN    1024
#define DDIM    128
#define NCHUNKS 112
#define INDIM   (NCHUNKS * 12 * DDIM)   // 172032
#define HIDDEN  512
#define NOUT    3
#define KSPLIT  32
#define KPER    (INDIM / KSPLIT)        // 5376, divisible by 4
#define NWIDTH  64                      // output columns per wave (4 WMMA tiles)

typedef float v2f __attribute__((ext_vector_type(2)));
typedef float v8f __attribute__((ext_vector_type(8)));

// ---------------------------------------------------------------------------
// Kernel 1: chunk features (mean, std, max, softmax-attn) for Z, dZ, ddZ.
// One workgroup per (chunk, batch). 128 threads = one per feature dim d.
// Writes flat feature matrix feat[b][c*1536 + sig*512 + f*128 + d].
// ---------------------------------------------------------------------------
__global__ __launch_bounds__(128) void feat_kernel(
    const float* __restrict__ Z,
    const float* __restrict__ qz,
    const float* __restrict__ qdz,
    const float* __restrict__ qddz,
    float* __restrict__ feat)
{
    __shared__ float smem[64 * 129];   // chunk tokens, padded stride (no bank conflicts)
    __shared__ float sc[64];           // raw attention scores
    __shared__ float wsm[64];          // softmax numerators

    const int c   = blockIdx.x;        // 0..111 global chunk id
    const int b   = blockIdx.y;        // 0..63
    const int tid = threadIdx.x;       // 0..127 == d

    int seg, t0;
    if (c < 64)      { seg = 16; t0 = c * 16; }
    else if (c < 96) { seg = 32; t0 = (c - 64) * 32; }
    else             { seg = 64; t0 = (c - 96) * 64; }

    const float* zb  = Z + (size_t)b * TLEN * DDIM;
    float*       out = feat + (size_t)b * INDIM + (size_t)c * 12 * DDIM;

    for (int sig = 0; sig < 3; ++sig) {
        const float* q = (sig == 0) ? qz : (sig == 1) ? qdz : qddz;

        float sum = 0.f, sq = 0.f, mx = -3.4e38f;
        // pass 1: materialize signal values into LDS, accumulate per-d stats
        for (int s = 0; s < seg; ++s) {
            const int t  = t0 + s;
            const float zc = zb[t * DDIM + tid];
            float x;
            if (sig == 0) {
                x = zc;
            } else {
                const float zp = (t >= 1) ? zb[(t - 1) * DDIM + tid] : 0.f;
                if (sig == 1) {
                    x = (t >= 1) ? (zc - zp) : 0.f;
                } else {
                    const float zpp = (t >= 2) ? zb[(t - 2) * DDIM + tid] : 0.f;
                    x = (t == 0) ? 0.f
                      : (t == 1) ? (zc - zp)
                      : (zc - 2.f * zp + zpp);
                }
            }
            smem[s * 129 + tid] = x;
            sum += x;
            sq  += x * x;
            mx   = fmaxf(mx, x);
        }
        __syncthreads();

        // pass 2: per-token dot with q  (thread s handles token s)
        if (tid < seg) {
            float dot = 0.f;
            #pragma unroll 4
            for (int d = 0; d < DDIM; ++d) dot += smem[tid * 129 + d] * q[d];
            sc[tid] = dot * 0.08838834764831845f;   // 1/sqrt(128)
        }
        __syncthreads();

        // softmax (each thread redundantly scans <=64 values; deterministic)
        float smax = -3.4e38f;
        for (int s = 0; s < seg; ++s) smax = fmaxf(smax, sc[s]);
        if (tid < seg) wsm[tid] = expf(sc[tid] - smax);
        __syncthreads();

        float den = 0.f;
        for (int s = 0; s < seg; ++s) den += wsm[s];
        float att = 0.f;
        for (int s = 0; s < seg; ++s) att += wsm[s] * smem[s * 129 + tid];
        att /= den;

        const float mean = sum / (float)seg;
        const float var  = sq / (float)seg - mean * mean;
        const float sd   = sqrtf(fmaxf(var, 0.f));

        out[sig * 512 + 0 * 128 + tid] = mean;
        out[sig * 512 + 1 * 128 + tid] = sd;
        out[sig * 512 + 2 * 128 + tid] = mx;
        out[sig * 512 + 3 * 128 + tid] = att;
        __syncthreads();   // before smem reuse for next signal
    }
}

// ---------------------------------------------------------------------------
// Kernel 2: fp32 WMMA GEMM  A[64][INDIM] x W1[INDIM][512] -> split-K partials.
// Grid (8 N-supertiles of 64 cols, 32 K-splits); block = 128 threads = 4
// wave32s; wave w owns rows [16w,16w+16) and ALL 64 columns of the supertile
// via 4 accumulators, so each A fragment feeds 4 WMMAs (4x less A traffic).
// A-frag per ISA 16x4 f32 layout: lane l holds A[m][k + 2*(l>>4) + {0,1}],
// m = l&15.  B-frag mirrored: lane l holds W1[k + 2*(l>>4) + {0,1}][n].
// W1 loaded non-temporally so the L2-resident feature matrix isn't evicted.
// Inner loop is pure pointer-increment: A ptr += 16B, B ptr += 8KB per step,
// all eight B loads as immediate offsets off one base (single clause).
// ---------------------------------------------------------------------------
__global__ __launch_bounds__(128) void gemm_kernel(
    const float* __restrict__ A,     // [64][INDIM]
    const float* __restrict__ Bw,    // [INDIM][512]
    float* __restrict__ part)        // [KSPLIT][64][512]
{
    const int lane = threadIdx.x & 31;
    const int wave = threadIdx.x >> 5;        // 0..3 -> M tile
    const int n0   = blockIdx.x * NWIDTH;
    const int kb   = blockIdx.y * KPER;

    const int m    = wave * 16 + (lane & 15);
    const int half = lane >> 4;               // 0 or 1

    // running pointers (incremented by constants inside the loop)
    const float* __restrict__ ap = A + (size_t)m * INDIM + kb + 2 * half;
    const float* __restrict__ bq = Bw + (size_t)(kb + 2 * half) * HIDDEN
                                      + n0 + (lane & 15);

    v8f acc0 = {}, acc1 = {}, acc2 = {}, acc3 = {};
    #pragma unroll 2
    for (int it = 0; it < KPER / 4; ++it) {
        v2f av;
        av.x = ap[0];
        av.y = ap[1];

        v2f b0, b1, b2, b3;
        b0.x = __builtin_nontemporal_load(&bq[0]);
        b0.y = __builtin_nontemporal_load(&bq[HIDDEN]);
        b1.x = __builtin_nontemporal_load(&bq[16]);
        b1.y = __builtin_nontemporal_load(&bq[HIDDEN + 16]);
        b2.x = __builtin_nontemporal_load(&bq[32]);
        b2.y = __builtin_nontemporal_load(&bq[HIDDEN + 32]);
        b3.x = __builtin_nontemporal_load(&bq[48]);
        b3.y = __builtin_nontemporal_load(&bq[HIDDEN + 48]);

        // (neg_a, A, neg_b, B, c_mod, C, reuse_a, reuse_b)
        acc0 = __builtin_amdgcn_wmma_f32_16x16x4_f32(
                   false, av, false, b0, (short)0, acc0, false, false);
        acc1 = __builtin_amdgcn_wmma_f32_16x16x4_f32(
                   false, av, false, b1, (short)0, acc1, false, false);
        acc2 = __builtin_amdgcn_wmma_f32_16x16x4_f32(
                   false, av, false, b2, (short)0, acc2, false, false);
        acc3 = __builtin_amdgcn_wmma_f32_16x16x4_f32(
                   false, av, false, b3, (short)0, acc3, false, false);

        ap += 4;                 // +16 bytes
        bq += 4 * HIDDEN;        // +8192 bytes
    }

    // C layout: acc[r] of lane l -> row (16*wave + r + 8*half), col tile + (l&15)
    float* pslab = part + (size_t)blockIdx.y * (BATCH * HIDDEN);
    const int row0 = wave * 16 + 8 * half;
    const int coll = lane & 15;
    #pragma unroll
    for (int r = 0; r < 8; ++r) {
        float* prow = pslab + (size_t)(row0 + r) * HIDDEN + n0 + coll;
        prow[0]  = acc0[r];
        prow[16] = acc1[r];
        prow[32] = acc2[r];
        prow[48] = acc3[r];
    }
}

// ---------------------------------------------------------------------------
// Kernel 3: reduce split-K partials, +b1, exact GELU, x W2 + b2, clip [0,1].
// One block per batch row.
// ---------------------------------------------------------------------------
__global__ __launch_bounds__(256) void mlp2_kernel(
    const float* __restrict__ part,   // [KSPLIT][64][512]
    const float* __restrict__ b1,
    const float* __restrict__ W2,     // [512][3]
    const float* __restrict__ b2,
    float* __restrict__ y)            // [64][3]
{
    __shared__ float hsm[HIDDEN];
    __shared__ float red[256 * 3];
    const int b = blockIdx.x;
    const int tid = threadIdx.x;

    for (int j = tid; j < HIDDEN; j += 256) {
        float v = b1[j];
        for (int ks = 0; ks < KSPLIT; ++ks)
            v += part[(size_t)ks * BATCH * HIDDEN + (size_t)b * HIDDEN + j];
        // exact GELU: 0.5*x*(1+erf(x/sqrt(2)))
        hsm[j] = 0.5f * v * (1.f + erff(v * 0.70710678118654752f));
    }
    __syncthreads();

    float a0 = 0.f, a1 = 0.f, a2 = 0.f;
    for (int j = tid; j < HIDDEN; j += 256) {
        const float h = hsm[j];
        a0 += h * W2[j * 3 + 0];
        a1 += h * W2[j * 3 + 1];
        a2 += h * W2[j * 3 + 2];
    }
    red[tid * 3 + 0] = a0;
    red[tid * 3 + 1] = a1;
    red[tid * 3 + 2] = a2;
    __syncthreads();

    if (tid < NOUT) {
        float s = b2[tid];
        for (int t = 0; t < 256; ++t) s += red[t * 3 + tid];
        y[b * NOUT + tid] = fminf(fmaxf(s, 0.f), 1.f);
    }
}

// ---------------------------------------------------------------------------
extern "C" void kernel_launch(void* const* d_in, const int* in_sizes, int n_in,
                              void* d_out, int out_size, void* d_ws, size_t ws_size,
                              hipStream_t stream)
{
    const float* Z    = (const float*)d_in[0];
    const float* qz   = (const float*)d_in[1];
    const float* qdz  = (const float*)d_in[2];
    const float* qddz = (const float*)d_in[3];
    const float* W1   = (const float*)d_in[4];
    const float* b1   = (const float*)d_in[5];
    const float* W2   = (const float*)d_in[6];
    const float* b2   = (const float*)d_in[7];
    float* out = (float*)d_out;

    float* feat = (float*)d_ws;                       // 64*172032 f32 (~44 MB)
    float* part = feat + (size_t)BATCH * INDIM;       // 32*64*512 f32 (4 MB)

    feat_kernel<<<dim3(NCHUNKS, BATCH), 128, 0, stream>>>(Z, qz, qdz, qddz, feat);
    gemm_kernel<<<dim3(HIDDEN / NWIDTH, KSPLIT), 128, 0, stream>>>(feat, W1, part);
    mlp2_kernel<<<BATCH, 256, 0, stream>>>(part, b1, W2, b2, out);
}